// EMG_88751204204652
// MI455X (gfx1250) — compile-verified
//
#include <hip/hip_runtime.h>

// EMG elementwise recurrent scan, MI455X / gfx1250.
// x: [B=8, S=2048, H=1024] f32 -> (output, mem, con) each [B, S, L=4, H] f32.
// 8192 independent (b,h) chains; each thread owns 2 adjacent h (float2).

#define B_NUM 8
#define S_LEN 2048
#define H_DIM 1024
#define L_NUM 4

typedef __attribute__((ext_vector_type(2))) float v2f;

__device__ __forceinline__ float tanh_fast(float x) {
#if __has_builtin(__builtin_amdgcn_tanhf)
  return __builtin_amdgcn_tanhf(x);          // v_tanh_f32 (gfx1250 TRANS op)
#else
  float r;
  asm("v_tanh_f32 %0, %1" : "=v"(r) : "v"(x));
  return r;
#endif
}

__device__ __forceinline__ v2f tanh2(v2f v) {
  v2f r;
  r.x = tanh_fast(v.x);
  r.y = tanh_fast(v.y);
  return r;
}

__global__ __launch_bounds__(32)
void emg_scan_kernel(const float* __restrict__ xg,
                     float* __restrict__ outg,
                     float* __restrict__ memg,
                     float* __restrict__ cong) {
  const int gid = blockIdx.x * 32 + threadIdx.x;   // 0..4095
  const int b   = gid >> 9;                        // H/2 = 512 threads per batch row
  const int h2  = (gid & 511) << 1;                // even h index

  // x[b][t][h2..h2+1]; advance by H_DIM floats (= H_DIM/2 v2f) per timestep.
  const v2f* __restrict__ xp =
      (const v2f*)(xg + (size_t)b * S_LEN * H_DIM + (size_t)h2);
  const size_t xs = H_DIM / 2;                     // v2f stride per timestep

  // out/mem/con index: ((b*S + t)*L + l)*H + h
  size_t ob = (size_t)b * S_LEN * L_NUM * H_DIM + (size_t)h2;
  const size_t os = (size_t)L_NUM * H_DIM;         // per-timestep stride (floats)

  v2f m[L_NUM], c[L_NUM];
#pragma unroll
  for (int l = 0; l < L_NUM; ++l) {
    v2f z = {0.f, 0.f};
    m[l] = z;
    c[l] = z;
  }

  // 4-deep register prefetch queue for x (read-once -> non-temporal loads).
  v2f xq0 = __builtin_nontemporal_load(xp + 0 * xs);
  v2f xq1 = __builtin_nontemporal_load(xp + 1 * xs);
  v2f xq2 = __builtin_nontemporal_load(xp + 2 * xs);
  v2f xq3 = __builtin_nontemporal_load(xp + 3 * xs);

  auto step = [&](v2f xv) {
    v2f li = xv;                                   // layer input chains through layers
    size_t o = ob;
#pragma unroll
    for (int l = 0; l < L_NUM; ++l) {
      v2f u  = li + m[l];                          // merges
      v2f v  = li + c[l];                          // context
      v2f iv = tanh2(u);                           // i = tanh(merges)
      v2f g  = tanh2(u * 0.5f) * 0.5f + 0.5f;      // sigmoid(merges)
      v2f mn = iv * g;                             // m_next
      v2f cn = tanh2(v) - iv;                      // c_next
      m[l] = mn;
      c[l] = cn;
      li = mn;

      __builtin_nontemporal_store(mn, (v2f*)(memg + o));
      __builtin_nontemporal_store(cn, (v2f*)(cong + o));
      v2f rl;
      rl.x = fmaxf(cn.x, 0.f);
      rl.y = fmaxf(cn.y, 0.f);
      __builtin_nontemporal_store(mn + rl, (v2f*)(outg + o));
      o += H_DIM;
    }
    ob += os;
  };

#pragma unroll 1
  for (int t0 = 0; t0 < S_LEN; t0 += 4) {
    // Warm GL2 ~32 timesteps ahead (global_prefetch_b8); register queue then
    // covers the L2->VGPR latency 4 timesteps ahead.
    if (t0 + 32 < S_LEN)
      __builtin_prefetch((const void*)(xp + (size_t)(t0 + 32) * xs), 0, 1);

    step(xq0);
    if (t0 + 4 < S_LEN) xq0 = __builtin_nontemporal_load(xp + (size_t)(t0 + 4) * xs);
    step(xq1);
    if (t0 + 5 < S_LEN) xq1 = __builtin_nontemporal_load(xp + (size_t)(t0 + 5) * xs);
    step(xq2);
    if (t0 + 6 < S_LEN) xq2 = __builtin_nontemporal_load(xp + (size_t)(t0 + 6) * xs);
    step(xq3);
    if (t0 + 7 < S_LEN) xq3 = __builtin_nontemporal_load(xp + (size_t)(t0 + 7) * xs);
  }
}

extern "C" void kernel_launch(void* const* d_in, const int* in_sizes, int n_in,
                              void* d_out, int out_size, void* d_ws, size_t ws_size,
                              hipStream_t stream) {
  (void)in_sizes; (void)n_in; (void)d_ws; (void)ws_size; (void)out_size;

  const float* x = (const float*)d_in[0];
  float* out = (float*)d_out;
  const size_t N = (size_t)B_NUM * S_LEN * L_NUM * H_DIM;   // 67,108,864
  float* mem = out + N;
  float* con = out + 2 * N;

  // 4096 threads total: 128 single-wave blocks spread across WGPs.
  dim3 grid((B_NUM * H_DIM / 2) / 32);   // 128
  dim3 block(32);
  emg_scan_kernel<<<grid, block, 0, stream>>>(x, out, mem, con);
}